// HungarianLoss_47493748359774
// MI455X (gfx1250) — compile-verified
//
#include <hip/hip_runtime.h>
#include <hip/hip_bf16.h>

#define Bsz 256
#define Nn  128
#define Dd  64
#define BIGF 1e9f

typedef __attribute__((ext_vector_type(2))) float v2f;
typedef __attribute__((ext_vector_type(8))) float v8f;

// ---------------------------------------------------------------------------
// Kernel 1: batched pairwise Euclidean distance via fp32 WMMA.
// One 16x16 output tile per wave; 8 waves (256 threads) per block cover one
// 16-row strip x all 128 cols of one batch. 16 chained v_wmma_f32_16x16x4_f32
// accumulate the K=64 dot products.
// ---------------------------------------------------------------------------
__global__ __launch_bounds__(256) void cdist_wmma_kernel(
    const float* __restrict__ set1, const float* __restrict__ set2,
    float* __restrict__ cost) {
  const int batch = blockIdx.x >> 3;
  const int tileM = (blockIdx.x & 7) << 4;
  const int tid   = threadIdx.x;
  const int wave  = tid >> 5;
  const int lane  = tid & 31;
  const int half  = lane >> 4;   // 0: K pair {0,1}; 1: K pair {2,3}
  const int lr    = lane & 15;
  const int tileN = wave << 4;

  const float* X = set1 + (size_t)batch * Nn * Dd;
  const float* Y = set2 + (size_t)batch * Nn * Dd;

  // Row squared-norms for this block's 16 X rows and all 128 Y rows.
  __shared__ float sx2[16];
  __shared__ float sy2[Nn];
  if (tid < 16) {
    const float* r = X + (size_t)(tileM + tid) * Dd;
    float s = 0.f;
#pragma unroll
    for (int k = 0; k < Dd; k += 4) {
      float4 v = *(const float4*)(r + k);
      s += v.x * v.x + v.y * v.y + v.z * v.z + v.w * v.w;
    }
    sx2[tid] = s;
  } else if (tid < 16 + Nn) {
    const int row = tid - 16;
    const float* r = Y + (size_t)row * Dd;
    float s = 0.f;
#pragma unroll
    for (int k = 0; k < Dd; k += 4) {
      float4 v = *(const float4*)(r + k);
      s += v.x * v.x + v.y * v.y + v.z * v.z + v.w * v.w;
    }
    sy2[row] = s;
  }

  // A lane L: x[tileM + L%16][k0 + 2*(L/16) + {0,1}]
  // B lane L: y[tileN + L%16][k0 + 2*(L/16) + {0,1}]
  const float* Arow = X + (size_t)(tileM + lr) * Dd + half * 2;
  const float* Brow = Y + (size_t)(tileN + lr) * Dd + half * 2;

  v8f acc = {};
#pragma unroll
  for (int k0 = 0; k0 < Dd; k0 += 4) {
    v2f a = *(const v2f*)(Arow + k0);
    v2f b = *(const v2f*)(Brow + k0);
    // 8 args: (neg_a, A, neg_b, B, c_mod, C, reuse_a, reuse_b)
    acc = __builtin_amdgcn_wmma_f32_16x16x4_f32(
        false, a, false, b, (short)0, acc, false, false);
  }
  __syncthreads();

  // D layout: lane L, VGPR r -> row M = r + 8*(L/16), col N = L%16
  const float y2c = sy2[tileN + lr];
  float* outcol = cost + (size_t)batch * Nn * Nn + tileN + lr;
#pragma unroll
  for (int r = 0; r < 8; ++r) {
    const int m = r + (half << 3);
    float sq = sx2[m] + y2c - 2.0f * acc[r];
    sq = fmaxf(sq, 1e-12f);
    outcol[(size_t)(tileM + m) * Nn] = sqrtf(sq);
  }
}

// ---------------------------------------------------------------------------
// Kernel 2: exact Jonker-Volgenant Hungarian per batch.
// One block (4 wave32s) per batch. Full 64KB cost matrix staged in LDS
// (CDNA5: 320KB LDS per WGP) via the gfx1250 async-copy path
// (global_load_async_to_lds_b128, ASYNCcnt) which bypasses VGPRs and
// overlaps the u/v/p initialization. Columns j in [0,128] strided across
// the 128 threads; argmin via wave32 shuffle with lowest-index tie-break
// (matches jnp.argmin) and a fixed-order cross-wave merge -> deterministic.
// ---------------------------------------------------------------------------
__global__ __launch_bounds__(128) void hungarian_kernel(
    const float* __restrict__ cost, float* __restrict__ partial) {
  const int n     = Nn;
  const int batch = blockIdx.x;
  const int tid   = threadIdx.x;
  const int lane  = tid & 31;
  const int wid   = tid >> 5;

  __shared__ float sCost[Nn * Nn];  // 64 KB
  __shared__ float sU[Nn + 1];
  __shared__ float sV[Nn + 1];
  __shared__ float sMinv[Nn + 1];
  __shared__ int   sP[Nn + 1];
  __shared__ int   sWay[Nn + 1];
  __shared__ int   sUsed[Nn + 1];
  __shared__ float sRedV[4];
  __shared__ int   sRedI[4];
  __shared__ int   sJ1;
  __shared__ float sDelta;

  // Stage the 64KB cost matrix into LDS with async global->LDS B128 copies:
  // 128 threads x 32 passes x 16B. LDS destination offset = low 32 bits of
  // the generic address of the __shared__ element (LDS aperture layout).
  {
    const char* gbase = (const char*)(cost + (size_t)batch * Nn * Nn);
#pragma unroll 4
    for (int idx = tid; idx < Nn * Nn / 4; idx += 128) {
      unsigned lds_off = (unsigned)(size_t)&sCost[idx * 4];
      const char* gaddr = gbase + (size_t)idx * 16;
      asm volatile("global_load_async_to_lds_b128 %0, %1, off"
                   :
                   : "v"(lds_off), "v"(gaddr)
                   : "memory");
    }
  }
  // Overlaps with the async DMA above.
  for (int j = tid; j <= n; j += 128) { sU[j] = 0.f; sV[j] = 0.f; sP[j] = 0; }
  asm volatile("s_wait_asynccnt 0" ::: "memory");
  __syncthreads();

  for (int i = 1; i <= n; ++i) {
    for (int j = tid; j <= n; j += 128) {
      sMinv[j] = BIGF; sUsed[j] = 0; sWay[j] = 0;
    }
    if (tid == 0) sP[0] = i;
    __syncthreads();

    int j0 = 0;
    while (true) {
      const int i0 = sP[j0];       // uniform
      if (i0 == 0) break;          // current column free -> augment
      if (tid == 0) sUsed[j0] = 1;
      __syncthreads();

      const float ui0  = sU[i0];
      const float* rowp = &sCost[(size_t)(i0 - 1) * Nn];  // a[i0][j]=rowp[j-1]

      float bestv = BIGF;
      int   bestj = 0;
      for (int j = tid; j <= n; j += 128) {
        const float aij = (j == 0) ? 0.0f : rowp[j - 1];
        const float cur = aij - ui0 - sV[j];
        const int used = sUsed[j];
        float mv = sMinv[j];
        if (!used && cur < mv) { mv = cur; sMinv[j] = mv; sWay[j] = j0; }
        const float masked = (used || j == 0) ? BIGF : mv;
        if (masked < bestv) { bestv = masked; bestj = j; }
      }
      // wave32 argmin with lowest-index tie-break
#pragma unroll
      for (int off = 16; off > 0; off >>= 1) {
        const float ov = __shfl_down(bestv, off, 32);
        const int   oj = __shfl_down(bestj, off, 32);
        if (ov < bestv || (ov == bestv && oj < bestj)) { bestv = ov; bestj = oj; }
      }
      if (lane == 0) { sRedV[wid] = bestv; sRedI[wid] = bestj; }
      __syncthreads();
      if (tid == 0) {
        float bv = sRedV[0]; int bj = sRedI[0];
#pragma unroll
        for (int w = 1; w < 4; ++w) {
          if (sRedV[w] < bv || (sRedV[w] == bv && sRedI[w] < bj)) {
            bv = sRedV[w]; bj = sRedI[w];
          }
        }
        sJ1 = bj; sDelta = bv;
      }
      __syncthreads();
      const int   j1    = sJ1;
      const float delta = sDelta;

      // Potential / slack updates (used columns map to distinct rows).
      for (int j = tid; j <= n; j += 128) {
        if (sUsed[j]) {
          sU[sP[j]] += delta;
          sV[j]     -= delta;
        } else {
          sMinv[j]  -= delta;
        }
      }
      __syncthreads();
      j0 = j1;
    }

    // Unwind augmenting path (serial, thread 0).
    if (tid == 0) {
      int jj = j0;
      while (jj != 0) { const int jn = sWay[jj]; sP[jj] = sP[jn]; jj = jn; }
    }
    __syncthreads();
  }

  // Matched sum: sum_j cost[p[j]-1][j-1]; fixed-order reduction.
  float s = 0.f;
  for (int j = 1 + tid; j <= n; j += 128)
    s += sCost[(size_t)(sP[j] - 1) * Nn + (j - 1)];
#pragma unroll
  for (int off = 16; off > 0; off >>= 1) s += __shfl_down(s, off, 32);
  if (lane == 0) sRedV[wid] = s;
  __syncthreads();
  if (tid == 0) partial[batch] = (sRedV[0] + sRedV[1]) + (sRedV[2] + sRedV[3]);
}

// ---------------------------------------------------------------------------
// Kernel 3: deterministic fixed-order final reduction -> scalar loss.
// ---------------------------------------------------------------------------
__global__ __launch_bounds__(256) void final_reduce_kernel(
    const float* __restrict__ partial, float* __restrict__ out) {
  __shared__ float s[Bsz];
  const int t = threadIdx.x;
  s[t] = partial[t];
  __syncthreads();
  for (int off = Bsz / 2; off > 0; off >>= 1) {
    if (t < off) s[t] += s[t + off];
    __syncthreads();
  }
  if (t == 0) out[0] = s[0] * (1.0f / ((float)Bsz * (float)Nn));
}

extern "C" void kernel_launch(void* const* d_in, const int* in_sizes, int n_in,
                              void* d_out, int out_size, void* d_ws, size_t ws_size,
                              hipStream_t stream) {
  const float* set1 = (const float*)d_in[0];  // [B, N, D] f32
  const float* set2 = (const float*)d_in[1];  // [B, N, D] f32
  float* cost    = (float*)d_ws;                         // B*N*N floats (16 MB)
  float* partial = cost + (size_t)Bsz * Nn * Nn;         // B floats

  cdist_wmma_kernel<<<dim3(Bsz * 8), dim3(256), 0, stream>>>(set1, set2, cost);
  hungarian_kernel<<<dim3(Bsz), dim3(128), 0, stream>>>(cost, partial);
  final_reduce_kernel<<<dim3(1), dim3(Bsz), 0, stream>>>(partial, (float*)d_out);
}